// ProxyAsWarpedTarget_81406810129101
// MI455X (gfx1250) — compile-verified
//
#include <hip/hip_runtime.h>
#include <hip/hip_bf16.h>
#include <stdint.h>

namespace {

constexpr int kB   = 16;
constexpr int kH   = 512;
constexpr int kW   = 512;
constexpr int kDH  = 256;
constexpr int kDW  = 256;
constexpr int kCOut = 5;

// ---------- gfx1250 helpers ----------

__device__ __forceinline__ void async_copy16(uint32_t lds_byte_addr, const void* gaddr) {
    // CDNA5 async global->LDS copy, 16 bytes per lane, tracked by ASYNCcnt.
    asm volatile("global_load_async_to_lds_b128 %0, %1, off"
                 :: "v"(lds_byte_addr), "v"(gaddr)
                 : "memory");
}

__device__ __forceinline__ void async_wait0() {
#if __has_builtin(__builtin_amdgcn_s_wait_asynccnt)
    __builtin_amdgcn_s_wait_asynccnt(0);
#else
    asm volatile("s_wait_asynccnt 0" ::: "memory");
#endif
}

__device__ __forceinline__ float fast_tanh(float x) {
#if __has_builtin(__builtin_amdgcn_tanhf)
    return __builtin_amdgcn_tanhf(x);
#elif __has_builtin(__builtin_amdgcn_tanh_f32)
    return __builtin_amdgcn_tanh_f32(x);
#else
    return tanhf(x);
#endif
}

__device__ __forceinline__ float fast_sigmoid(float x) {
    return 1.0f / (1.0f + __expf(-x));
}

// ---------- Kernel 1: antialiased 2x downsample (jax.image.resize bilinear) ----------
// Separable triangle filter, taps {2i-1..2i+2} with weights {1,3,3,1}/8,
// edge-renormalized over valid taps (matches jax's weight normalization).

__global__ void __launch_bounds__(256)
downsample_rows_kernel(const float* __restrict__ src, float* __restrict__ dst) {
    __shared__ __align__(16) float rows[4 * kW];  // 8 KB: 4 source rows

    const int tid = threadIdx.x;          // 0..255 == output column j
    const int i   = blockIdx.x % kDH;     // output row
    const int bc  = blockIdx.x / kDH;     // b*5 + c
    const int r0  = 2 * i - 1;            // first vertical tap (may be -1)

    // Stage 4 input rows (clamped addresses; invalid taps get zero weight below).
    const uint32_t lds0 = (uint32_t)(uintptr_t)(&rows[0]);
    {
        int chunk = tid;                                  // chunks 0..255 -> rows 0,1
        int k = chunk >> 7;                               // 128 x 16B chunks per row
        int r = min(max(r0 + k, 0), kH - 1);
        const char* g = (const char*)(src + ((size_t)bc * kH + r) * kW)
                        + (size_t)(chunk & 127) * 16;
        async_copy16(lds0 + (uint32_t)chunk * 16u, g);

        chunk = tid + 256;                                // chunks 256..511 -> rows 2,3
        k = chunk >> 7;
        r = min(max(r0 + k, 0), kH - 1);
        g = (const char*)(src + ((size_t)bc * kH + r) * kW)
            + (size_t)(chunk & 127) * 16;
        async_copy16(lds0 + (uint32_t)chunk * 16u, g);
    }
    async_wait0();
    __syncthreads();

    const float wt[4] = {0.125f, 0.375f, 0.375f, 0.125f};
    const int j  = tid;
    const int c0 = 2 * j - 1;

    float whs = 0.0f;
    #pragma unroll
    for (int a = 0; a < 4; ++a)
        whs += ((c0 + a) >= 0 && (c0 + a) < kW) ? wt[a] : 0.0f;

    float acc = 0.0f, wvs = 0.0f;
    #pragma unroll
    for (int k = 0; k < 4; ++k) {
        const int r = r0 + k;
        const float wv = (r >= 0 && r < kH) ? wt[k] : 0.0f;
        wvs += wv;
        float ra = 0.0f;
        #pragma unroll
        for (int a = 0; a < 4; ++a) {
            const int c = c0 + a;
            const float wh = (c >= 0 && c < kW) ? wt[a] : 0.0f;
            const int cc = min(max(c, 0), kW - 1);
            ra += wh * rows[k * kW + cc];
        }
        acc += wv * ra;
    }
    dst[(size_t)bc * (kDH * kDW) + (size_t)i * kDW + j] = acc / (wvs * whs);
}

// Fallback: compute one downsampled value straight from global (L2-resident).
__device__ float downsample_point(const float* __restrict__ src, int bc, int i, int j) {
    const float wt[4] = {0.125f, 0.375f, 0.375f, 0.125f};
    const int r0 = 2 * i - 1, c0 = 2 * j - 1;
    float whs = 0.0f;
    #pragma unroll
    for (int a = 0; a < 4; ++a)
        whs += ((c0 + a) >= 0 && (c0 + a) < kW) ? wt[a] : 0.0f;
    float acc = 0.0f, wvs = 0.0f;
    #pragma unroll
    for (int k = 0; k < 4; ++k) {
        const int r = r0 + k;
        const float wv = (r >= 0 && r < kH) ? wt[k] : 0.0f;
        wvs += wv;
        const int rc = min(max(r, 0), kH - 1);
        const float* rp = src + ((size_t)bc * kH + rc) * kW;
        float ra = 0.0f;
        #pragma unroll
        for (int a = 0; a < 4; ++a) {
            const int c = c0 + a;
            const float wh = (c >= 0 && c < kW) ? wt[a] : 0.0f;
            const int cc = min(max(c, 0), kW - 1);
            ra += wh * rp[cc];
        }
        acc += wv * ra;
    }
    return acc / (wvs * whs);
}

// ---------- Kernel 2: upsample + activations + grid_sample + outputs ----------

template <bool FROM_WS>
__global__ void __launch_bounds__(256)
fuse_kernel(const float* __restrict__ dsrc,     // ws (FROM_WS) or raw `out`
            const float* __restrict__ target,
            float* __restrict__ proxy,
            float* __restrict__ gcout,
            float* __restrict__ addout) {
    const int idx = blockIdx.x * 256 + threadIdx.x;   // exact cover: 16*512*512
    const int x = idx & (kW - 1);
    const int y = (idx >> 9) & (kH - 1);
    const int b = idx >> 18;

    // 2x bilinear upsample: sample pos 0.5*x - 0.25; clamping == jax edge renorm.
    const int jx = (x - 1) >> 1;
    const int jy = (y - 1) >> 1;
    const float fx = (x & 1) ? 0.25f : 0.75f;
    const float fy = (y & 1) ? 0.25f : 0.75f;
    const int x0 = max(jx, 0),     x1 = min(jx + 1, kDW - 1);
    const int y0 = max(jy, 0),     y1 = min(jy + 1, kDH - 1);

    float o[kCOut];
    #pragma unroll
    for (int c = 0; c < kCOut; ++c) {
        const int bc = b * kCOut + c;
        float v00, v01, v10, v11;
        if (FROM_WS) {
            const float* p = dsrc + (size_t)bc * (kDH * kDW);
            v00 = p[y0 * kDW + x0]; v01 = p[y0 * kDW + x1];
            v10 = p[y1 * kDW + x0]; v11 = p[y1 * kDW + x1];
        } else {
            v00 = downsample_point(dsrc, bc, y0, x0);
            v01 = downsample_point(dsrc, bc, y0, x1);
            v10 = downsample_point(dsrc, bc, y1, x0);
            v11 = downsample_point(dsrc, bc, y1, x1);
        }
        const float top = (1.0f - fx) * v00 + fx * v01;
        const float bot = (1.0f - fx) * v10 + fx * v11;
        o[c] = (1.0f - fy) * top + fy * bot;
    }

    const float gcx = 0.05f * fast_tanh(o[0]);
    const float gcy = 0.05f * fast_tanh(o[1]);

    // base grid: linspace(-1,1,512)
    float gxn = fmaf((float)x, 2.0f / 511.0f, -1.0f) + gcx;
    float gyn = fmaf((float)y, 2.0f / 511.0f, -1.0f) + gcy;
    gxn = fminf(fmaxf(gxn, -1.0f), 1.0f);
    gyn = fminf(fmaxf(gyn, -1.0f), 1.0f);

    const float gx = fmaf(gxn + 1.0f, (float)kW * 0.5f, -0.5f);
    const float gy = fmaf(gyn + 1.0f, (float)kH * 0.5f, -0.5f);

    const float x0f = floorf(gx), y0f = floorf(gy);
    const float wx1 = gx - x0f,   wy1 = gy - y0f;
    const float wx0 = 1.0f - wx1, wy0 = 1.0f - wy1;
    const int ix0 = (int)x0f, iy0 = (int)y0f;
    const int ix1 = ix0 + 1,  iy1 = iy0 + 1;
    const float mx0 = (ix0 >= 0 && ix0 < kW) ? 1.0f : 0.0f;
    const float mx1 = (ix1 >= 0 && ix1 < kW) ? 1.0f : 0.0f;
    const float my0 = (iy0 >= 0 && iy0 < kH) ? 1.0f : 0.0f;
    const float my1 = (iy1 >= 0 && iy1 < kH) ? 1.0f : 0.0f;
    const int cx0 = min(max(ix0, 0), kW - 1);
    const int cx1 = min(max(ix1, 0), kW - 1);
    const int cy0 = min(max(iy0, 0), kH - 1);
    const int cy1 = min(max(iy1, 0), kH - 1);
    const float w00 = wx0 * wy0 * mx0 * my0;
    const float w01 = wx1 * wy0 * mx1 * my0;
    const float w10 = wx0 * wy1 * mx0 * my1;
    const float w11 = wx1 * wy1 * mx1 * my1;

    const float mean_[3] = {0.485f, 0.456f, 0.406f};
    const float rstd_[3] = {1.0f / 0.229f, 1.0f / 0.224f, 1.0f / 0.225f};

    const size_t hw  = (size_t)kH * kW;
    const float* tb  = target + (size_t)b * 3 * hw;
    const size_t pix = (size_t)y * kW + x;

    #pragma unroll
    for (int c = 0; c < 3; ++c) {
        const float* tc = tb + (size_t)c * hw;
        const float s = w00 * tc[(size_t)cy0 * kW + cx0]
                      + w01 * tc[(size_t)cy0 * kW + cx1]
                      + w10 * tc[(size_t)cy1 * kW + cx0]
                      + w11 * tc[(size_t)cy1 * kW + cx1];
        const float add = 0.1f * (fast_sigmoid(o[2 + c]) - mean_[c]) * rstd_[c];
        const size_t oidx = (size_t)(b * 3 + c) * hw + pix;
        proxy[oidx]  = s + add;
        addout[oidx] = add;
    }
    const size_t gidx = ((size_t)b * hw + pix) * 2;
    gcout[gidx + 0] = gcx;
    gcout[gidx + 1] = gcy;
}

} // anonymous namespace

extern "C" void kernel_launch(void* const* d_in, const int* in_sizes, int n_in,
                              void* d_out, int out_size, void* d_ws, size_t ws_size,
                              hipStream_t stream) {
    const float* net    = (const float*)d_in[0];  // (16,5,512,512) fp32
    const float* target = (const float*)d_in[1];  // (16,3,512,512) fp32

    float* outp = (float*)d_out;
    const size_t proxyN = (size_t)kB * 3 * kH * kW;   // 12,582,912
    const size_t gcN    = (size_t)kB * kH * kW * 2;   //  8,388,608
    float* proxy  = outp;
    float* gcout  = outp + proxyN;
    float* addout = outp + proxyN + gcN;

    const size_t wsNeed = (size_t)kB * kCOut * kDH * kDW * sizeof(float); // ~21 MB
    const dim3 blk(256);
    const int fuseBlocks = (kB * kH * kW) / 256;      // 16384

    if (ws_size >= wsNeed) {
        float* ws = (float*)d_ws;
        downsample_rows_kernel<<<kB * kCOut * kDH, blk, 0, stream>>>(net, ws);
        fuse_kernel<true><<<fuseBlocks, blk, 0, stream>>>(ws, target, proxy, gcout, addout);
    } else {
        fuse_kernel<false><<<fuseBlocks, blk, 0, stream>>>(net, target, proxy, gcout, addout);
    }
}